// get_edge_feature_3040836845599
// MI455X (gfx1250) — compile-verified
//
#include <hip/hip_runtime.h>

#define BATCH 2
#define NP    8192
#define ND    64
#define KNN   17   // K_NEIGHBORS + 1 (self included)

typedef __attribute__((ext_vector_type(2))) float v2f;
typedef __attribute__((ext_vector_type(8))) float v8f;

// ---------------------------------------------------------------------------
// Kernel 0: per-point squared norms  sq[b][n] = sum_d pc[b][d][n]^2
// ---------------------------------------------------------------------------
__global__ __launch_bounds__(256)
void sq_kernel(const float* __restrict__ pc, float* __restrict__ sq)
{
    int t = blockIdx.x * 256 + threadIdx.x;        // 0 .. BATCH*NP-1
    int b = t / NP, n = t % NP;
    const float* P = pc + (size_t)b * ND * NP + n;
    float s = 0.f;
#pragma unroll 8
    for (int d = 0; d < ND; ++d) { float v = P[(size_t)d * NP]; s += v * v; }
    sq[t] = s;
}

// ---------------------------------------------------------------------------
// Kernel 1: fused distance GEMM (f32 WMMA 16x16x4) + streaming top-17 select.
// One wave (32 lanes) owns a 16-query tile; loops over all 512 candidate
// tiles of 16. knn_idx layout: [B][N][KNN], ascending distance.
// ---------------------------------------------------------------------------
__global__ __launch_bounds__(256)
void knn_kernel(const float* __restrict__ pc, const float* __restrict__ sq,
                int* __restrict__ knn_idx)
{
    __shared__ float tileD[8][16 * 17];     // per-wave 16x16 dist tile (+pad)
    __shared__ float mergeD[8][32 * KNN];   // per-wave lane lists (dist)
    __shared__ int   mergeI[8][32 * KNN];   // per-wave lane lists (idx)

    const int lane = threadIdx.x & 31;
    const int wv   = threadIdx.x >> 5;
    const int w    = blockIdx.x * 8 + wv;       // 0 .. BATCH*(NP/16)-1
    const int b     = w >> 9;                    // / (NP/16)
    const int qbase = (w & 511) << 4;

    const float* P  = pc + (size_t)b * ND * NP;
    const float* SQ = sq + (size_t)b * NP;

    const int hi  = lane >> 4;                   // lane group (0/1)
    const int col = lane & 15;

    // A fragments: 16x4 f32 per k-step; lane layout per CDNA5 ISA:
    //   lanes 0-15: M=lane, K = k0+{0,1}; lanes 16-31: M=lane-16, K = k0+{2,3}
    v2f A[16];
#pragma unroll
    for (int kk = 0; kk < 16; ++kk) {
        int k0 = kk * 4 + 2 * hi;
        A[kk].x = P[(size_t)k0       * NP + qbase + col];
        A[kk].y = P[(size_t)(k0 + 1) * NP + qbase + col];
    }

    // per-lane query norms for the 8 accumulator rows (vgpr r -> M = r + 8*hi)
    float sqq[8];
#pragma unroll
    for (int r = 0; r < 8; ++r) sqq[r] = SQ[qbase + r + 8 * hi];

    // Register-resident sorted top-17: lane handles row (lane>>1),
    // column half (lane&1)*8 of every 16-wide candidate tile.
    float ld[KNN]; int li[KNN];
#pragma unroll
    for (int j = 0; j < KNN; ++j) { ld[j] = 3.0e38f; li[j] = 0; }
    const int myrow = lane >> 1;
    const int chalf = (lane & 1) * 8;

    float* tl = tileD[wv];

    for (int ct = 0; ct < NP / 16; ++ct) {
        const int cbase = ct * 16;

        // hint next candidate tile into cache (speculative, L2-resident anyway)
        __builtin_prefetch(&P[(size_t)(2 * hi) * NP + cbase + 16 + col], 0, 3);

        v8f acc = {0.f, 0.f, 0.f, 0.f, 0.f, 0.f, 0.f, 0.f};
#pragma unroll
        for (int kk = 0; kk < 16; ++kk) {
            int k0 = kk * 4 + 2 * hi;
            v2f Bf;
            Bf.x = P[(size_t)k0       * NP + cbase + col];
            Bf.y = P[(size_t)(k0 + 1) * NP + cbase + col];
            // D = A x B + C, full f32 precision (16x16x4)
            acc = __builtin_amdgcn_wmma_f32_16x16x4_f32(
                false, A[kk], false, Bf, (short)0, acc, false, false);
        }

        const float sqc = SQ[cbase + col];
        // C/D layout: vgpr r, lanes 0-15 -> M=r, N=lane; lanes 16-31 -> M=r+8
#pragma unroll
        for (int r = 0; r < 8; ++r) {
            float d2 = sqq[r] + sqc - 2.0f * acc[r];
            tl[(r + 8 * hi) * 17 + col] = d2;
        }
        __builtin_amdgcn_wave_barrier();

        // streaming insert: 8 candidates per lane, branchless bubble insert
#pragma unroll
        for (int c = 0; c < 8; ++c) {
            float d  = tl[myrow * 17 + chalf + c];
            int   id = cbase + chalf + c;
            if (__any(d < ld[KNN - 1])) {
                float cd = d; int ci = id;
#pragma unroll
                for (int j = 0; j < KNN; ++j) {
                    bool sw = cd < ld[j];             // strict: earlier idx wins ties
                    float nd = sw ? cd : ld[j];
                    float xd = sw ? ld[j] : cd;
                    int   ni = sw ? ci : li[j];
                    int   xi = sw ? li[j] : ci;
                    ld[j] = nd; li[j] = ni; cd = xd; ci = xi;
                }
            }
        }
        __builtin_amdgcn_wave_barrier();
    }

    // merge the two half-row lists (lane pair) via LDS
    float* mD = mergeD[wv]; int* mI = mergeI[wv];
#pragma unroll
    for (int j = 0; j < KNN; ++j) { mD[lane * KNN + j] = ld[j]; mI[lane * KNN + j] = li[j]; }
    __builtin_amdgcn_wave_barrier();

    if ((lane & 1) == 0) {
        int p0 = lane * KNN, p1 = p0 + KNN;
        int i0 = 0, i1 = 0;
        int row = qbase + (lane >> 1);
        int* outp = knn_idx + ((size_t)b * NP + row) * KNN;
        for (int j = 0; j < KNN; ++j) {              // i0+i1 == j <= 16: in-bounds
            float d0 = mD[p0 + i0], d1 = mD[p1 + i1];
            int   x0 = mI[p0 + i0], x1 = mI[p1 + i1];
            bool take0 = (d0 < d1) || (d0 == d1 && x0 < x1);
            outp[j] = take0 ? x0 : x1;
            if (take0) ++i0; else ++i1;
        }
    }
}

// ---------------------------------------------------------------------------
// Kernel 2: edge features. out layout: edge [B][2D][KNN][NP] then idx (float)
// [B][KNN][NP]. Thread per n -> coalesced stores; neighbor gathers hit L2
// (input per batch is 2 MB vs 192 MB L2).
// ---------------------------------------------------------------------------
__global__ __launch_bounds__(256)
void edge_kernel(const float* __restrict__ pc, const int* __restrict__ knn,
                 float* __restrict__ out)
{
    int n = blockIdx.x * 256 + threadIdx.x;
    int k = blockIdx.y;
    int b = blockIdx.z;

    int nbr = knn[((size_t)b * NP + n) * KNN + k];
    const float* P = pc + (size_t)b * ND * NP;
    float* O = out + ((size_t)b * 2 * ND * KNN + (size_t)k) * NP + n;

#pragma unroll 4
    for (int d = 0; d < ND; ++d) {
        float cen = P[(size_t)d * NP + n];
        float nb  = P[(size_t)d * NP + nbr];
        O[(size_t)d * KNN * NP]        = cen;       // central, broadcast over K
        O[(size_t)(d + ND) * KNN * NP] = nb - cen;  // neighbor - central
    }
    const size_t EDGE = (size_t)BATCH * 2 * ND * KNN * NP;
    out[EDGE + ((size_t)b * KNN + k) * NP + n] = (float)nbr;
}

// ---------------------------------------------------------------------------
extern "C" void kernel_launch(void* const* d_in, const int* in_sizes, int n_in,
                              void* d_out, int out_size, void* d_ws, size_t ws_size,
                              hipStream_t stream)
{
    (void)in_sizes; (void)n_in; (void)out_size; (void)ws_size;
    const float* pc = (const float*)d_in[0];

    float* sq  = (float*)d_ws;                                          // 64 KB
    int*   knn = (int*)((char*)d_ws + (size_t)BATCH * NP * sizeof(float)); // 1.06 MB
    float* out = (float*)d_out;

    sq_kernel  <<<dim3((BATCH * NP) / 256), dim3(256), 0, stream>>>(pc, sq);
    knn_kernel <<<dim3(BATCH * (NP / 16) / 8), dim3(256), 0, stream>>>(pc, sq, knn);
    edge_kernel<<<dim3(NP / 256, KNN, BATCH), dim3(256), 0, stream>>>(pc, knn, out);
}